// DeltaEncoder_317827580116
// MI455X (gfx1250) — compile-verified
//
#include <hip/hip_runtime.h>
#include <hip/hip_bf16.h>
#include <stdint.h>

typedef float v4f __attribute__((ext_vector_type(4)));

#define B_ 16
#define T_ 2048
#define C_ 32
#define O_ 64
#define N_ (B_ * C_ * T_) /* 1048576 */

// ---------------------------------------------------------------------------
// Kernel 1: per-block partial sums of delta and delta^2 (deterministic).
// x layout [B,T,C]; linear i = (b*T + t)*C + c, so lane-consecutive i is
// coalesced. delta[t==0] = 0.
// ---------------------------------------------------------------------------
__global__ __launch_bounds__(256) void k_stats(const float* __restrict__ x,
                                               float* __restrict__ part) {
    float s = 0.f, sq = 0.f;
    const int stride = gridDim.x * blockDim.x;
    for (int i = blockIdx.x * blockDim.x + threadIdx.x; i < N_; i += stride) {
        const int t = (i >> 5) & (T_ - 1);          // i = (b*T + t)*C + c, C=32
        const float xi = x[i];
        const float d = (t == 0) ? 0.f : (xi - x[i - C_]);
        s += d;
        sq += d * d;
    }
    __shared__ float rs[256], rq[256];
    rs[threadIdx.x] = s; rq[threadIdx.x] = sq;
    __syncthreads();
    for (int off = 128; off > 0; off >>= 1) {
        if (threadIdx.x < off) {
            rs[threadIdx.x] += rs[threadIdx.x + off];
            rq[threadIdx.x] += rq[threadIdx.x + off];
        }
        __syncthreads();
    }
    if (threadIdx.x == 0) {
        part[2 * blockIdx.x]     = rs[0];
        part[2 * blockIdx.x + 1] = rq[0];
    }
}

// ---------------------------------------------------------------------------
// Kernel 2: reduce 256 partials -> scale = gamma*rsqrt(var+eps),
//                                  shift = beta - mean*scale
// ---------------------------------------------------------------------------
__global__ __launch_bounds__(256) void k_finalize(const float* __restrict__ part,
                                                  const float* __restrict__ bnw,
                                                  const float* __restrict__ bnb,
                                                  float* __restrict__ sc) {
    __shared__ float rs[256], rq[256];
    rs[threadIdx.x] = part[2 * threadIdx.x];
    rq[threadIdx.x] = part[2 * threadIdx.x + 1];
    __syncthreads();
    for (int off = 128; off > 0; off >>= 1) {
        if (threadIdx.x < off) {
            rs[threadIdx.x] += rs[threadIdx.x + off];
            rq[threadIdx.x] += rq[threadIdx.x + off];
        }
        __syncthreads();
    }
    if (threadIdx.x == 0) {
        const float invN  = 1.0f / (float)N_;
        const float mean  = rs[0] * invN;
        const float var   = rq[0] * invN - mean * mean;
        const float scale = bnw[0] * rsqrtf(var + 1e-5f);
        sc[0] = scale;
        sc[1] = bnb[0] - mean * scale;
    }
}

// ---------------------------------------------------------------------------
// Kernel 3: main. Block = 256 threads; each thread owns one (b,c) and 4
// consecutive t values -> coalesced b128 NT stores along T (the dominant
// 256 MB of traffic, the whole roofline). W/b staged into LDS with the
// CDNA5 async load-to-LDS path. The LIF recurrence over o is sequential;
// 4 independent chains per thread give ILP to cover the store stream.
// ---------------------------------------------------------------------------
__global__ __launch_bounds__(256) void k_main(const float* __restrict__ x,
                                              const float* __restrict__ W,
                                              const float* __restrict__ bvec,
                                              const float* __restrict__ sc,
                                              float* __restrict__ out) {
    __shared__ float sW[O_];
    __shared__ float sB[O_];

    // Wave 0 (lanes 0..31) pulls W[64] and b[64] into LDS: one b64/lane each,
    // via the gfx1250 async-to-LDS path (tracked on ASYNCcnt).
    if (threadIdx.x < 32) {
        const uint32_t lane = threadIdx.x;
        const uint32_t ow = (uint32_t)(uintptr_t)(&sW[0]) + lane * 8u;
        const uint32_t ob = (uint32_t)(uintptr_t)(&sB[0]) + lane * 8u;
        const float* gw = W + 2u * lane;
        const float* gb = bvec + 2u * lane;
        asm volatile("global_load_async_to_lds_b64 %0, %1, off"
                     :: "v"(ow), "v"(gw) : "memory");
        asm volatile("global_load_async_to_lds_b64 %0, %1, off"
                     :: "v"(ob), "v"(gb) : "memory");
        asm volatile("s_wait_asynccnt 0x0" ::: "memory");
    }
    __syncthreads();

    const float scale = sc[0];
    const float shift = sc[1];

    const int bc = blockIdx.x >> 1;                       // b*C + c
    const int t0 = (blockIdx.x & 1) * 1024 + threadIdx.x * 4;
    const int b  = bc >> 5;
    const int c  = bc & 31;

    // x[b, t0-1 .. t0+3, c]; lane-strided 128B-stride reads are absorbed by
    // L2 (x is only 4 MB; each line is reused by 32 different c-blocks).
    const float* xp = x + ((size_t)b * T_ + (size_t)t0) * C_ + c;
    const float x0 = xp[0];
    const float x1 = xp[C_];
    const float x2 = xp[2 * C_];
    const float x3 = xp[3 * C_];
    const float xm1 = *(xp - (t0 ? C_ : 0));   // t0==0 -> xm1=x0 -> delta=0

    const float d0 = (x0 - xm1) * scale + shift;
    const float d1 = (x1 - x0) * scale + shift;
    const float d2 = (x2 - x1) * scale + shift;
    const float d3 = (x3 - x2) * scale + shift;

    float v0 = 0.f, v1 = 0.f, v2 = 0.f, v3 = 0.f;
    float* op = out + (size_t)bc * T_ + (size_t)t0;

#pragma unroll 4
    for (int o = 0; o < O_; ++o) {
        const float wo = sW[o];
        const float bo = sB[o];

        // Match reference rounding: v_h = v + (enc - v) * 0.5  (TAU = 2, /2 exact)
        const float e0 = fmaf(d0, wo, bo) - v0;
        const float e1 = fmaf(d1, wo, bo) - v1;
        const float e2 = fmaf(d2, wo, bo) - v2;
        const float e3 = fmaf(d3, wo, bo) - v3;
        const float vh0 = fmaf(e0, 0.5f, v0);
        const float vh1 = fmaf(e1, 0.5f, v1);
        const float vh2 = fmaf(e2, 0.5f, v2);
        const float vh3 = fmaf(e3, 0.5f, v3);

        const bool s0 = vh0 >= 1.0f;
        const bool s1 = vh1 >= 1.0f;
        const bool s2 = vh2 >= 1.0f;
        const bool s3 = vh3 >= 1.0f;

        v4f r;
        r.x = s0 ? 1.0f : 0.0f;
        r.y = s1 ? 1.0f : 0.0f;
        r.z = s2 ? 1.0f : 0.0f;
        r.w = s3 ? 1.0f : 0.0f;

        v0 = s0 ? 0.0f : vh0;
        v1 = s1 ? 0.0f : vh1;
        v2 = s2 ? 0.0f : vh2;
        v3 = s3 ? 0.0f : vh3;

        // Stream 256 MB of spikes past L2 (output > 192 MB L2; never re-read)
        __builtin_nontemporal_store(r, (v4f*)(op + (size_t)o * N_));
    }
}

// ---------------------------------------------------------------------------
extern "C" void kernel_launch(void* const* d_in, const int* in_sizes, int n_in,
                              void* d_out, int out_size, void* d_ws, size_t ws_size,
                              hipStream_t stream) {
    const float* x   = (const float*)d_in[0];
    const float* bnw = (const float*)d_in[1];
    const float* bnb = (const float*)d_in[2];
    const float* W   = (const float*)d_in[3];
    const float* bb  = (const float*)d_in[4];
    float* out  = (float*)d_out;
    float* part = (float*)d_ws;        // 512 floats of partials
    float* sc   = part + 512;          // {scale, shift}

    k_stats   <<<256,  256, 0, stream>>>(x, part);
    k_finalize<<<1,    256, 0, stream>>>(part, bnw, bnb, sc);
    k_main    <<<1024, 256, 0, stream>>>(x, W, bb, sc, out);
}